// Candemann_Parafac_module_73993696575955
// MI455X (gfx1250) — compile-verified
//
#include <hip/hip_runtime.h>

// CDNA5 / gfx1250: wave32, WMMA f32 16x16x4 (native fp32 matrix op).
typedef __attribute__((ext_vector_type(2))) float v2f;
typedef __attribute__((ext_vector_type(8))) float v8f;

#define K_SLICES 7
#define NMAT     64
#define WAVES_PER_SLICE 4                    // four 16-row slabs per 64x64 slice
#define NWAVES   (K_SLICES * WAVES_PER_SLICE) // 28 waves
#define NTHREADS (NWAVES * 32)                // 896 threads, one workgroup

__global__ __launch_bounds__(NTHREADS)
void masked_trisum_wmma(const float* __restrict__ x,
                        const float* __restrict__ beta0,
                        float* __restrict__ out) {
    __shared__ float partials[NWAVES];

    const int tid  = threadIdx.x;
    const int wave = tid >> 5;        // 0..27
    const int lane = tid & 31;
    const int k    = wave >> 2;       // 0..6  : slice of x
    const int trow = wave & 3;        // 0..3  : 16-row slab
    const int m    = lane & 15;       // row within slab (A-matrix M index)
    const int half = lane >> 4;       // lanes 0-15 -> K=0,1 ; lanes 16-31 -> K=2,3
    const int row  = trow * 16 + m;   // global row index i

    const float* xs = x + k * (NMAT * NMAT) + row * NMAT;

    v8f acc = {};                     // C/D accumulator (16x16 f32, 8 VGPRs)
    v2f bones; bones[0] = 1.0f; bones[1] = 1.0f;   // B = ones(4x16)

    // Walk 64 columns in chunks of 4: D += A(16x4) * ones(4x16).
    // Every column of D accumulates the masked row-sum of this slab.
#pragma unroll
    for (int c = 0; c < NMAT; c += 4) {
        const int col0 = c + half * 2;
        float a0 = xs[col0];          // contiguous pair -> global_load_b64
        float a1 = xs[col0 + 1];
        // strictly-lower mask: include x[i,j] iff j < i (per-lane select, no EXEC divergence)
        a0 = (col0     < row) ? a0 : 0.0f;
        a1 = (col0 + 1 < row) ? a1 : 0.0f;
        v2f a; a[0] = a0; a[1] = a1;
        // 8-arg form: (neg_a, A, neg_b, B, c_mod, C, reuse_a, reuse_b)
        acc = __builtin_amdgcn_wmma_f32_16x16x4_f32(
            false, a, false, bones, (short)0, acc, false, false);
    }

    // D layout: lane n (n<16) holds M=0..7 for column n; lane n+16 holds M=8..15.
    // All 16 columns are identical, so (lane 0 + lane 16) sums = slab total.
    float s = acc[0] + acc[1] + acc[2] + acc[3] + acc[4] + acc[5] + acc[6] + acc[7];
    s += __shfl_xor(s, 16, 32);

    if (lane == 0) partials[wave] = s;
    __syncthreads();

    if (tid == 0) {
        float total = 0.0f;
#pragma unroll
        for (int w = 0; w < NWAVES; ++w) total += partials[w];
        const float cp_sum = 8.0f * (0.2f * 0.2f * 0.2f);   // RANK * 0.2^3
        out[0] = beta0[0] + cp_sum * total;
    }
}

extern "C" void kernel_launch(void* const* d_in, const int* in_sizes, int n_in,
                              void* d_out, int out_size, void* d_ws, size_t ws_size,
                              hipStream_t stream) {
    const float* x     = (const float*)d_in[0];   // (7, 64, 64) fp32
    const float* beta0 = (const float*)d_in[1];   // scalar fp32
    float* out = (float*)d_out;                   // 1 fp32
    hipLaunchKernelGGL(masked_trisum_wmma, dim3(1), dim3(NTHREADS), 0, stream,
                       x, beta0, out);
}